// NearestNeighbor_77747497992211
// MI455X (gfx1250) — compile-verified
//
#include <hip/hip_runtime.h>
#include <hip/hip_bf16.h>
#include <float.h>

// ---------------------------------------------------------------------------
// Problem constants (from reference): K=5, C=3(+wall)=4, H=W=48, T_rec=16,
// x = (1, 3, 3, 48, 48).  FEAT = 4*25 = 100.  NSRC = 15*2304 = 34560.
// ---------------------------------------------------------------------------
#define HH    48
#define WW    48
#define HWPX  (HH * WW)        // 2304
#define CCH   4
#define KWIN  5
#define FEAT  (CCH * KWIN * KWIN)   // 100
#define NSRC  (15 * HWPX)      // 34560 candidate rows
#define NQ    HWPX             // 2304 query rows
#define MTILES (NSRC / 16)     // 2160 candidate tiles (even)
#define MPAIRS (MTILES / 2)    // 1080 tile-pairs
#define NTILES (NQ / 16)       // 144
#define KSLICES 25             // K=4 slices covering FEAT=100
#define KPAIRS  13             // 13 slice-pairs (26th slice = zero pad)

typedef __attribute__((ext_vector_type(2))) float v2f;
typedef __attribute__((ext_vector_type(4))) float v4f;
typedef __attribute__((ext_vector_type(8))) float v8f;

// ---------------------------------------------------------------------------
// Packed B layout, matched to V_WMMA_F32_16X16X4_F32 operand registers:
// candidate tile mt (16 candidates), slice-pair kp, lane L (wave32):
//   float index = ((mt*13 + kp)*32 + L)*4 + q
//   q=0,1: slice 2kp   -> feature d = 8kp     + 2*(L>>4) + q
//   q=2,3: slice 2kp+1 -> feature d = 8kp + 4 + 2*(L>>4) + (q-2)
//   candidate m = mt*16 + (L&15)
// One global_load_b128 per lane per kp => 13 coalesced loads per tile.
// Pair kp=12 second slice (kk=25, d=100..103) is zero padding.
// ---------------------------------------------------------------------------
__global__ void prep_src(const float* __restrict__ rec,
                         const float* __restrict__ walls,
                         float* __restrict__ srcPk,
                         float* __restrict__ halfNorm) {
  int m = blockIdx.x * blockDim.x + threadIdx.x;
  if (m >= NSRC) return;
  int t = m / HWPX, p = m - t * HWPX;
  int y = p / WW, x = p - y * WW;
  const int mt = m >> 4, l16 = m & 15;
  float nrm = 0.0f;
  for (int c = 0; c < CCH; ++c) {
    const float* img = (c < 3) ? (rec + (t * 3 + c) * HWPX) : walls;
    for (int i = 0; i < KWIN; ++i) {
      int yy = y + i - 2; if (yy < 0) yy += HH; else if (yy >= HH) yy -= HH;
      for (int j = 0; j < KWIN; ++j) {
        int xx = x + j - 2; if (xx < 0) xx += WW; else if (xx >= WW) xx -= WW;
        float v = img[yy * WW + xx];
        nrm += v * v;
        const int d  = c * 25 + i * 5 + j;       // feature 0..99
        const int kk = d >> 2, r = d & 3;        // slice, pos-in-slice
        const int kp = kk >> 1, sub = kk & 1;    // pair, slice-in-pair
        const int L  = l16 + ((r >> 1) << 4);    // lane (half from r)
        const int q  = (sub << 1) | (r & 1);
        srcPk[(((size_t)mt * KPAIRS + kp) * 32 + L) * 4 + q] = v;
      }
    }
  }
  // zero-pad slice kk=25 (kp=12, sub=1)
  for (int r = 0; r < 4; ++r) {
    const int L = l16 + ((r >> 1) << 4);
    const int q = 2 | (r & 1);
    srcPk[(((size_t)mt * KPAIRS + 12) * 32 + L) * 4 + q] = 0.0f;
  }
  halfNorm[m] = 0.5f * nrm;
}

// ---------------------------------------------------------------------------
// Build transposed query table QT[d][n] from a 4-channel carry image
// (ch012 = 3*2304 floats, ch3 = 2304 floats).
// ---------------------------------------------------------------------------
__global__ void prep_query(const float* __restrict__ ch012,
                           const float* __restrict__ ch3,
                           float* __restrict__ QT) {
  int n = blockIdx.x * blockDim.x + threadIdx.x;
  if (n >= NQ) return;
  int y = n / WW, x = n - y * WW;
  for (int c = 0; c < CCH; ++c) {
    const float* img = (c < 3) ? (ch012 + c * HWPX) : ch3;
    for (int i = 0; i < KWIN; ++i) {
      int yy = y + i - 2; if (yy < 0) yy += HH; else if (yy >= HH) yy -= HH;
      for (int j = 0; j < KWIN; ++j) {
        int xx = x + j - 2; if (xx < 0) xx += WW; else if (xx >= WW) xx -= WW;
        QT[(c * 25 + i * 5 + j) * NQ + n] = img[yy * WW + xx];
      }
    }
  }
}

// Copy frame 0 (first 3 channels of x) to output and zero the loss accumulator.
__global__ void init_out(const float* __restrict__ x,
                         float* __restrict__ out,
                         float* __restrict__ lossAcc) {
  int i = blockIdx.x * blockDim.x + threadIdx.x;
  if (i < 3 * HWPX) out[i] = x[i];
  if (i == 0) *lossAcc = 0.0f;
}

// ---------------------------------------------------------------------------
// GEMM + argmin.  One block per 16-query tile (256 threads = 8 waves).
// Each wave processes TWO candidate tiles per iteration with two interleaved
// accumulator chains sharing the resident A registers: 26 coalesced b128
// loads feeding 52 v_wmma_f32_16x16x4_f32 (2 independent RAW chains).
// Score 0.5||s||^2 - q.s; per-register running argmin (D layout: lane ->
// col N, reg r -> row M=r+8*half); 16-lane shfl_xor butterfly, then 8-wave
// combine in LDS; winner gathers the matched pixel's next-frame center.
// ---------------------------------------------------------------------------
__global__ __launch_bounds__(256) void gemm_argmin(
    const float* __restrict__ QT, const float* __restrict__ srcPk,
    const float* __restrict__ halfNorm,
    const float* __restrict__ rec, const float* __restrict__ walls,
    const float* __restrict__ targ012,   // target frame, 3 channels
    float* __restrict__ outFrame,        // output frame, 3 channels
    float* __restrict__ carryNext,       // 4-channel carry for next step
    float* __restrict__ lossAcc) {
  __shared__ float s_val[8 * 16];
  __shared__ int   s_idx[8 * 16];

  const int lane = threadIdx.x & 31;
  const int wave = threadIdx.x >> 5;
  const int half = lane >> 4;     // 0: lanes 0-15, 1: lanes 16-31
  const int l16  = lane & 15;
  const int n0   = blockIdx.x * 16;

  // A tile: 16 queries x 100 features, 26 slices (last = zero pad), 2 regs
  // per slice.  Layout (ISA 7.12.2, 32-bit A 16x4): M=l16, reg j -> K=j+2*half.
  float a0[2 * KPAIRS], a1[2 * KPAIRS];
  {
    const int col = n0 + l16;
#pragma unroll
    for (int kk = 0; kk < KSLICES; ++kk) {
      const int d = 4 * kk + 2 * half;
      a0[kk] = QT[d * NQ + col];
      a1[kk] = QT[(d + 1) * NQ + col];
    }
    a0[25] = 0.0f; a1[25] = 0.0f;
  }

  float bestV[8];
  int   bestI[8];
#pragma unroll
  for (int r = 0; r < 8; ++r) { bestV[r] = FLT_MAX; bestI[r] = 0; }

  const v4f* __restrict__ Bp4 = (const v4f*)srcPk;

  for (int pr = wave; pr < MPAIRS; pr += 8) {   // 1080/8 = 135 iterations
    const int mt0 = 2 * pr;                     // two adjacent candidate tiles
    const int base0 = mt0 * (KPAIRS * 32) + lane;       // float4 units
    const int base1 = base0 + (KPAIRS * 32);
    // Unconditional speculative prefetch of this wave's next pair (address
    // stays inside the workspace allocation even on the final iterations).
    __builtin_prefetch(&Bp4[base0 + 16 * (KPAIRS * 32)], 0, 0);

    const float hn0 = halfNorm[mt0 * 16 + l16];
    const float hn1 = halfNorm[mt0 * 16 + 16 + l16];

    v8f acc0 = {0.f, 0.f, 0.f, 0.f, 0.f, 0.f, 0.f, 0.f};
    v8f acc1 = {0.f, 0.f, 0.f, 0.f, 0.f, 0.f, 0.f, 0.f};
#pragma unroll
    for (int kp = 0; kp < KPAIRS; ++kp) {
      const v4f bq0 = Bp4[base0 + kp * 32];
      const v4f bq1 = Bp4[base1 + kp * 32];
      v2f a, b;
      a.x = a0[2 * kp];     a.y = a1[2 * kp];
      b.x = bq0.x;          b.y = bq0.y;
      acc0 = __builtin_amdgcn_wmma_f32_16x16x4_f32(
          false, a, false, b, (short)0, acc0, false, false);
      b.x = bq1.x;          b.y = bq1.y;
      acc1 = __builtin_amdgcn_wmma_f32_16x16x4_f32(
          false, a, false, b, (short)0, acc1, false, false);
      a.x = a0[2 * kp + 1]; a.y = a1[2 * kp + 1];
      b.x = bq0.z;          b.y = bq0.w;
      acc0 = __builtin_amdgcn_wmma_f32_16x16x4_f32(
          false, a, false, b, (short)0, acc0, false, false);
      b.x = bq1.z;          b.y = bq1.w;
      acc1 = __builtin_amdgcn_wmma_f32_16x16x4_f32(
          false, a, false, b, (short)0, acc1, false, false);
    }

    const int mcol0 = mt0 * 16 + l16;
    const int mcol1 = mcol0 + 16;
#pragma unroll
    for (int r = 0; r < 8; ++r) {          // reg r -> query M = r + 8*half
      const float s0 = hn0 - acc0[r];
      if (s0 < bestV[r]) { bestV[r] = s0; bestI[r] = mcol0; }
      const float s1 = hn1 - acc1[r];
      if (s1 < bestV[r]) { bestV[r] = s1; bestI[r] = mcol1; }
    }
  }

  // Reduce across the 16 lanes of each half-wave (candidates striped on N),
  // then park per-wave winners in LDS.
#pragma unroll
  for (int r = 0; r < 8; ++r) {
    float v = bestV[r]; int ix = bestI[r];
#pragma unroll
    for (int off = 8; off >= 1; off >>= 1) {
      float ov = __shfl_xor(v, off, 32);
      int   oi = __shfl_xor(ix, off, 32);
      if (ov < v || (ov == v && oi < ix)) { v = ov; ix = oi; }
    }
    if (l16 == 0) {
      s_val[wave * 16 + r + 8 * half] = v;
      s_idx[wave * 16 + r + 8 * half] = ix;
    }
  }
  __syncthreads();

  // Combine 8 waves; winner gathers the matched pixel of the NEXT frame.
  if (threadIdx.x < 16) {
    const int M = threadIdx.x;
    float v = s_val[M]; int ix = s_idx[M];
    for (int w = 1; w < 8; ++w) {
      float ov = s_val[w * 16 + M]; int oi = s_idx[w * 16 + M];
      if (ov < v || (ov == v && oi < ix)) { v = ov; ix = oi; }
    }
    const int t = ix / HWPX, p = ix - t * HWPX;   // matched frame/pixel
    const int n = n0 + M;
    float lloss = 0.0f;
#pragma unroll
    for (int c = 0; c < 3; ++c) {
      const float g = rec[((t + 1) * 3 + c) * HWPX + p];
      outFrame[c * HWPX + n]  = g;
      carryNext[c * HWPX + n] = g;
      const float d = g - targ012[c * HWPX + n];
      lloss += d * d;
    }
    const float gw = walls[p];                // wall channel of matched patch
    carryNext[3 * HWPX + n] = gw;
    const float dw = gw - walls[n];           // target wall channel = walls
    lloss += dw * dw;
    atomicAdd(lossAcc, lloss);
  }
}

__global__ void finalize(const float* __restrict__ lossAcc,
                         float* __restrict__ out) {
  if (blockIdx.x == 0 && threadIdx.x == 0)
    out[3 * 3 * HWPX] = *lossAcc * (1.0f / (2.0f * 4.0f * HWPX));
}

// ---------------------------------------------------------------------------
extern "C" void kernel_launch(void* const* d_in, const int* in_sizes, int n_in,
                              void* d_out, int out_size, void* d_ws, size_t ws_size,
                              hipStream_t stream) {
  (void)in_sizes; (void)n_in; (void)out_size; (void)ws_size;
  const float* x         = (const float*)d_in[0];  // (1,3,3,48,48)
  const float* recording = (const float*)d_in[1];  // (16,3,48,48)
  const float* walls     = (const float*)d_in[2];  // (48,48)
  float* out = (float*)d_out;                      // 3*3*2304 frames + 1 loss

  float* ws       = (float*)d_ws;
  float* srcPk    = ws;                                  // 2160*13*128 floats
  float* halfNorm = srcPk + (size_t)MTILES * KPAIRS * 128;  // 34560
  float* QT       = halfNorm + NSRC;                     // 100*2304
  float* carry    = QT + (size_t)FEAT * NQ;              // 4*2304
  float* lossAcc  = carry + 4 * HWPX;                    // 1

  prep_src<<<(NSRC + 255) / 256, 256, 0, stream>>>(recording, walls, srcPk, halfNorm);
  init_out<<<(3 * HWPX + 255) / 256, 256, 0, stream>>>(x, out, lossAcc);

  // Step 0: query = x frame 0 (+ walls), target = x frame 1.
  prep_query<<<(NQ + 255) / 256, 256, 0, stream>>>(x, walls, QT);
  gemm_argmin<<<NTILES, 256, 0, stream>>>(QT, srcPk, halfNorm, recording, walls,
                                          x + 3 * HWPX, out + 3 * HWPX,
                                          carry, lossAcc);

  // Step 1: query = step-0 result (4-channel carry), target = x frame 2.
  prep_query<<<(NQ + 255) / 256, 256, 0, stream>>>(carry, carry + 3 * HWPX, QT);
  gemm_argmin<<<NTILES, 256, 0, stream>>>(QT, srcPk, halfNorm, recording, walls,
                                          x + 6 * HWPX, out + 6 * HWPX,
                                          carry, lossAcc);

  finalize<<<1, 32, 0, stream>>>(lossAcc, out);
}